// HierarchicalBernoulliLDS_44281112822395
// MI455X (gfx1250) — compile-verified
//
#include <hip/hip_runtime.h>
#include <hip/hip_bf16.h>
#include <math.h>

typedef __attribute__((ext_vector_type(2))) float v2f;
typedef __attribute__((ext_vector_type(8))) float v8f;

#define S_SUBJ 512
#define T_TRIAL 8192
#define D_IN 8

// ---------- inverse normal CDF (Acklam), |rel err| ~1e-9, far beyond f32 ----------
__device__ __forceinline__ double ndtri_acklam(double p) {
    const double a1=-3.969683028665376e+01, a2= 2.209460984245205e+02, a3=-2.759285104469687e+02,
                 a4= 1.383577518672690e+02, a5=-3.066479806614716e+01, a6= 2.506628277459239e+00;
    const double b1=-5.447609879822406e+01, b2= 1.615858368580409e+02, b3=-1.556989798598866e+02,
                 b4= 6.680131188771972e+01, b5=-1.328068155288572e+01;
    const double c1=-7.784894002430293e-03, c2=-3.223964580411365e-01, c3=-2.400758277161838e+00,
                 c4=-2.549732539343734e+00, c5= 4.374664141464968e+00, c6= 2.938163982698783e+00;
    const double d1= 7.784695709041462e-03, d2= 3.224671290700398e-01,
                 d3= 2.445134137142996e+00, d4= 3.754408661907416e+00;
    const double plow = 0.02425, phigh = 1.0 - plow;
    double q, r, x;
    if (p < plow) {
        q = sqrt(-2.0*log(p));
        x = (((((c1*q+c2)*q+c3)*q+c4)*q+c5)*q+c6) / ((((d1*q+d2)*q+d3)*q+d4)*q+1.0);
    } else if (p <= phigh) {
        q = p - 0.5; r = q*q;
        x = (((((a1*r+a2)*r+a3)*r+a4)*r+a5)*r+a6)*q / (((((b1*r+b2)*r+b3)*r+b4)*r+b5)*r+1.0);
    } else {
        q = sqrt(-2.0*log(1.0-p));
        x = -(((((c1*q+c2)*q+c3)*q+c4)*q+c5)*q+c6) / ((((d1*q+d2)*q+d3)*q+d4)*q+1.0);
    }
    return x;
}

// ---------- Kernel 1: parallel AR(1) scan, one wave32 per subject ----------
__global__ __launch_bounds__(32)
void ar1_scan_kernel(const float* __restrict__ u_a, const float* __restrict__ sigmasq,
                     const float* __restrict__ eps_mu, const float* __restrict__ eps_x0,
                     const float* __restrict__ eps_x, float* __restrict__ x_out) {
    __shared__ float buf[T_TRIAL];                 // 32 KB of 320 KB LDS
    const int s    = blockIdx.x;
    const int lane = threadIdx.x;                  // wave32

    // Per-subject params (uniform across lanes; cheap double math once)
    const double SQRT1_2 = 0.70710678118654752440;
    const double Fa = 0.5 * erfc( 9.9 * SQRT1_2);  // Phi((0-0.99)/0.1) = Phi(-9.9)
    const double Fb = 0.5 * erfc(-0.1 * SQRT1_2);  // Phi((1-0.99)/0.1) = Phi(0.1)
    const float  a  = (float)(0.99 + 0.1 * ndtri_acklam(Fa + (double)u_a[s] * (Fb - Fa)));
    const float  mu = eps_mu[s];
    const float  b  = mu * (1.0f - a);
    const float  x0 = mu + eps_x0[s];              // sqrt(SIGMASQ0)=1
    const float  sd = sqrtf(sigmasq[s]);

    // Stage eps into LDS, coalesced
    const float* ep = eps_x + (size_t)s * T_TRIAL;
    for (int i = lane; i < T_TRIAL; i += 32) buf[i] = ep[i];
    __syncthreads();

    const int CH = T_TRIAL / 32;                   // 256 steps/lane
    const int c0 = lane * CH;

    // Phase 1: reduce chunk to affine transform x -> P*x + r, P = a^CH
    float r = 0.0f;
    #pragma unroll 8
    for (int j = 0; j < CH; ++j)
        r = fmaf(a, r, fmaf(sd, buf[c0 + j], b));
    float P = a;
    #pragma unroll
    for (int i = 0; i < 8; ++i) P *= P;            // a^256

    // Inclusive composition scan across lanes: (P2,R2)o(P1,R1) = (P2*P1, P2*R1+R2)
    float Pc = P, Rc = r;
    #pragma unroll
    for (int off = 1; off < 32; off <<= 1) {
        float Pp = __shfl_up(Pc, off);
        float Rp = __shfl_up(Rc, off);
        if (lane >= off) { Rc = fmaf(Pc, Rp, Rc); Pc = Pc * Pp; }
    }
    // Exclusive: x at chunk start
    float Pe = __shfl_up(Pc, 1);
    float Re = __shfl_up(Rc, 1);
    float x  = (lane == 0) ? x0 : fmaf(Pe, x0, Re);

    // Phase 2: materialize x_t (collected BEFORE each update), in-place over eps
    #pragma unroll 8
    for (int j = 0; j < CH; ++j) {
        float e = buf[c0 + j];
        buf[c0 + j] = x;                           // x[s, c0+j]
        x = fmaf(a, x, fmaf(sd, e, b));            // x[t+1] = a*x[t] + b + sd*eps[t]
    }
    __syncthreads();

    // Coalesced write-out
    float* xo = x_out + (size_t)s * T_TRIAL;
    for (int i = lane; i < T_TRIAL; i += 32) xo[i] = buf[i];
}

// ---------- Kernel 2: logits via f32 WMMA (16x16x4), sigmoid-Bernoulli ----------
__global__ __launch_bounds__(256)
void logit_wmma_kernel(const float* __restrict__ inputs, const float* __restrict__ w_all,
                       const float* __restrict__ x_all, const float* __restrict__ u_bern,
                       int* __restrict__ y_out) {
    const int s    = blockIdx.y;
    const int lane = threadIdx.x & 31;
    const int wave = threadIdx.x >> 5;
    const int half = lane >> 4;                    // 0: lanes 0-15, 1: lanes 16-31
    const int n    = lane & 15;                    // N / t index within tile
    const int dlo  = 2 * half;                     // A/B K-pair held by this half-wave
    const int tbase = blockIdx.x * 512 + wave * 64;

    // A = w replicated across M rows. 32-bit 16x4 A layout:
    // lanes 0-15: {K=0,K=1}; lanes 16-31: {K=2,K=3}
    const float* wp = w_all + (size_t)s * D_IN;    // w = eps_w (MU_W=0, SIGMA_W=1)
    v2f A0 = *reinterpret_cast<const v2f*>(wp + dlo);       // K = 0..3
    v2f A1 = *reinterpret_cast<const v2f*>(wp + 4 + dlo);   // K = 4..7

    const float* inb = inputs + (size_t)s * T_TRIAL * D_IN;

    #pragma unroll
    for (int it = 0; it < 4; ++it) {
        const int t0 = tbase + it * 16;
        // B[k][n] = inputs[s, t0+n, kb+k]; lane holds {kb+dlo, kb+dlo+1} for column n
        const float* bp = inb + (size_t)(t0 + n) * D_IN + dlo;
        v2f B0 = *reinterpret_cast<const v2f*>(bp);         // kb = 0
        v2f B1 = *reinterpret_cast<const v2f*>(bp + 4);     // kb = 4
        __builtin_prefetch(bp + 16 * D_IN, 0, 0);           // next tile -> global_prefetch_b8

        v8f c = {};
        c = __builtin_amdgcn_wmma_f32_16x16x4_f32(false, A0, false, B0, (short)0, c, false, false);
        c = __builtin_amdgcn_wmma_f32_16x16x4_f32(false, A1, false, B1, (short)0, c, false, false);
        // All M rows identical => c[0] = dot(inputs[s,t0+n,:], w)

        if (half == 0) {                                    // one copy per t
            const size_t idx = (size_t)s * T_TRIAL + (t0 + n);
            const float logit = x_all[idx] + c[0];
            // y = (u < 1/(1+e^-logit))  <=>  u*(1+e^-logit) < 1   (since 1+e > 0)
            // division-free: avoids the v_div_scale/v_div_fmas chain
            const float e = expf(-logit);
            y_out[idx] = (fmaf(u_bern[idx], e, u_bern[idx]) < 1.0f) ? 1 : 0;
        }
    }
}

extern "C" void kernel_launch(void* const* d_in, const int* in_sizes, int n_in,
                              void* d_out, int out_size, void* d_ws, size_t ws_size,
                              hipStream_t stream) {
    const float* inputs  = (const float*)d_in[0];  // [S,T,D]
    const float* eps_w   = (const float*)d_in[1];  // [S,D]  (== w since MU_W=0, SIGMA_W=1)
    const float* u_a     = (const float*)d_in[2];  // [S]
    const float* sigmasq = (const float*)d_in[3];  // [S]
    const float* eps_mu  = (const float*)d_in[4];  // [S]
    const float* eps_x0  = (const float*)d_in[5];  // [S]
    const float* eps_x   = (const float*)d_in[6];  // [S,T]
    const float* u_bern  = (const float*)d_in[7];  // [S,T]

    float* x_out = (float*)d_out;                                    // first S*T floats
    int*   y_out = (int*)((float*)d_out + (size_t)S_SUBJ * T_TRIAL); // next S*T int32

    ar1_scan_kernel<<<S_SUBJ, 32, 0, stream>>>(u_a, sigmasq, eps_mu, eps_x0, eps_x, x_out);

    dim3 grid(T_TRIAL / 512, S_SUBJ);
    logit_wmma_kernel<<<grid, 256, 0, stream>>>(inputs, eps_w, x_out, u_bern, y_out);
}